// HardVoxelization_3985729651011
// MI455X (gfx1250) — compile-verified
//
#include <hip/hip_runtime.h>
#include <hip/hip_bf16.h>
#include <stdint.h>

// ---- problem constants (match reference) ----
#define GRIDX 1408
#define GRIDY 1600
#define GRIDZ 40
#define NVOX_LIN (GRIDX * GRIDY * GRIDZ)          // 90,112,000  (< 2^31)
#define NWORDS ((NVOX_LIN + 31) / 32)             // 2,816,000 bitmap words
#define SCAN_BS 256
#define NBLK ((NWORDS + SCAN_BS - 1) / SCAN_BS)   // 11,000 scan blocks
#define MAX_VOX 120000
#define MAX_PTS 32
#define EMPTY_SLOT 0xFFFFFFFFu
#define INVALID_LIN 0xFFFFFFFFu                   // > any valid lin (max 90,111,999)

// ---- workspace layout (uint32 units) ----
static constexpr unsigned OFF_BITMAP = 0u;
static constexpr unsigned OFF_WPRE   = (unsigned)NWORDS;                 // exclusive prefix of word popcounts
static constexpr unsigned OFF_BSUM   = 2u * (unsigned)NWORDS;            // per-block sums (NBLK, padded)
static constexpr unsigned OFF_COUNT  = 2u * (unsigned)NWORDS + 11008u;   // per-voxel total counts
static constexpr unsigned OFF_LINR   = OFF_COUNT + (unsigned)MAX_VOX;    // rank -> linear voxel id
static constexpr unsigned OFF_SLOT   = OFF_LINR + (unsigned)MAX_VOX;     // [MAX_VOX][32] point-index slots
static constexpr unsigned OFF_TOTAL  = OFF_SLOT + (unsigned)(MAX_VOX * MAX_PTS); // scalar
static constexpr unsigned OFF_PLIN   = OFF_TOTAL + 16u;                  // per-point cached lin [n]

// ------------------------------------------------------- B128 memset
__global__ void vox_memset_u128(uint4* __restrict__ p, unsigned val, unsigned n4) {
    unsigned i = blockIdx.x * blockDim.x + threadIdx.x;
    unsigned stride = gridDim.x * blockDim.x;
    uint4 v = make_uint4(val, val, val, val);
    for (; i < n4; i += stride) p[i] = v;          // global_store_b128
}

// ---------------------------------------------------------- voxel id calc
__device__ __forceinline__ bool vox_lin_calc(const float4 p, unsigned& lin) {
    // identical fp32 math to reference: floor((p - lo) / voxel), cast int32
    float fx = floorf((p.x - 0.0f)  / 0.05f);
    float fy = floorf((p.y + 40.0f) / 0.05f);
    float fz = floorf((p.z + 3.0f)  / 0.1f);
    int ix = (int)fx, iy = (int)fy, iz = (int)fz;
    if (ix < 0 || ix >= GRIDX || iy < 0 || iy >= GRIDY || iz < 0 || iz >= GRIDZ) return false;
    lin = (unsigned)ix * (unsigned)(GRIDY * GRIDZ) + (unsigned)iy * (unsigned)GRIDZ + (unsigned)iz;
    return true;
}

// --------------------- pass 1: mark bitmap + cache per-point linear id
__global__ void vox_mark_kernel(const float4* __restrict__ pts,
                                unsigned* __restrict__ bitmap,
                                unsigned* __restrict__ plin, int n) {
    int i = blockIdx.x * blockDim.x + threadIdx.x;
    if (i >= n) return;
    // gfx1250 global_prefetch_b8: pull the stream ahead into GL2
    __builtin_prefetch(pts + i + 4096, 0, 1);
    float4 p = pts[i];               // global_load_b128
    unsigned lin;
    if (!vox_lin_calc(p, lin)) {
        plin[i] = INVALID_LIN;
        return;
    }
    plin[i] = lin;
    atomicOr(&bitmap[lin >> 5], 1u << (lin & 31u));
}

// ------------------------------- pass 2a: block-local scan of popcounts
// Bitmap words staged global -> LDS via CDNA5 async-tensor path
// (GLOBAL_LOAD_ASYNC_TO_LDS_B32, ASYNCcnt), then scanned in LDS.
__global__ void vox_scan_block(const unsigned* __restrict__ bitmap,
                               unsigned* __restrict__ wpre,
                               unsigned* __restrict__ bsum) {
    __shared__ unsigned stage[SCAN_BS];
    __shared__ unsigned sh[SCAN_BS];
    unsigned i = blockIdx.x * SCAN_BS + threadIdx.x;
    unsigned c = 0u;
    if (i < (unsigned)NWORDS) {
#if defined(__gfx1250__)
        // low 32 bits of a generic pointer to LDS == wave-relative LDS byte offset
        unsigned lds_off = (unsigned)(size_t)(&stage[threadIdx.x]);
        unsigned long long gaddr = (unsigned long long)(size_t)(bitmap + i);
        asm volatile("global_load_async_to_lds_b32 %0, %1, off"
                     :: "v"(lds_off), "v"(gaddr)
                     : "memory");
        asm volatile("s_wait_asynccnt 0x0" ::: "memory");
        c = (unsigned)__popc(stage[threadIdx.x]);
#else
        c = (unsigned)__popc(bitmap[i]);
#endif
    }
    sh[threadIdx.x] = c;
    __syncthreads();
    unsigned incl = c;
    for (unsigned off = 1; off < SCAN_BS; off <<= 1) {
        unsigned t = (threadIdx.x >= off) ? sh[threadIdx.x - off] : 0u;
        __syncthreads();
        incl += t;
        sh[threadIdx.x] = incl;
        __syncthreads();
    }
    if (i < (unsigned)NWORDS) wpre[i] = incl - c;            // exclusive within block
    if (threadIdx.x == SCAN_BS - 1) bsum[blockIdx.x] = incl; // block total
}

// ----------------------- pass 2b: single-block scan of the block sums
__global__ void vox_scan_bsum(unsigned* __restrict__ bsum,
                              unsigned* __restrict__ totalp, int nblk) {
    __shared__ unsigned sh[SCAN_BS];
    __shared__ unsigned runningSh;
    if (threadIdx.x == 0) runningSh = 0u;
    __syncthreads();
    for (int base = 0; base < nblk; base += SCAN_BS) {
        int i = base + (int)threadIdx.x;
        unsigned v = (i < nblk) ? bsum[i] : 0u;
        sh[threadIdx.x] = v;
        __syncthreads();
        unsigned incl = v;
        for (unsigned off = 1; off < SCAN_BS; off <<= 1) {
            unsigned t = (threadIdx.x >= off) ? sh[threadIdx.x - off] : 0u;
            __syncthreads();
            incl += t;
            sh[threadIdx.x] = incl;
            __syncthreads();
        }
        unsigned running = runningSh;
        if (i < nblk) bsum[i] = running + incl - v;           // global exclusive
        __syncthreads();
        if (threadIdx.x == SCAN_BS - 1) runningSh = running + incl;
        __syncthreads();
    }
    if (threadIdx.x == 0) *totalp = runningSh;                // total present voxels
}

// --------------------------- pass 2c: add block offsets into word prefix
__global__ void vox_add_offsets(unsigned* __restrict__ wpre,
                                const unsigned* __restrict__ bsum) {
    unsigned i = blockIdx.x * blockDim.x + threadIdx.x;
    if (i < (unsigned)NWORDS) wpre[i] += bsum[i >> 8];        // SCAN_BS == 256
}

// ----- pass 3: rank lookup + deterministic 32-min atomicMin slot insert
// Reads the cached per-point lin (b32) instead of re-reading 32 MB of points.
__global__ void vox_insert_kernel(const unsigned* __restrict__ plin,
                                  const unsigned* __restrict__ bitmap,
                                  const unsigned* __restrict__ wpre,
                                  unsigned* __restrict__ counters,
                                  unsigned* __restrict__ linOfRank,
                                  unsigned* __restrict__ slots, int n) {
    int i = blockIdx.x * blockDim.x + threadIdx.x;
    if (i >= n) return;
    __builtin_prefetch(plin + i + 16384, 0, 1);
    unsigned lin = plin[i];
    if (lin == INVALID_LIN) return;
    unsigned w = lin >> 5, b = lin & 31u;
    // rank == position of lin in sorted unique ids == searchsorted(uniq, lin)
    unsigned rank = wpre[w] + (unsigned)__popc(bitmap[w] & ((1u << b) - 1u));
    if (rank >= (unsigned)MAX_VOX) return;                    // beyond first-120000 voxels
    linOfRank[rank] = lin;                                    // idempotent
    atomicAdd(&counters[rank], 1u);
    // insertion cascade: final slot state = 32 smallest point indices, ascending
    // (== reference's stable-order first-32 points). Deterministic.
    unsigned cur = (unsigned)i;
    unsigned* sl = slots + rank * MAX_PTS;
#pragma unroll 1
    for (int s = 0; s < MAX_PTS; ++s) {
        unsigned prev = atomicMin(&sl[s], cur);
        if (prev == EMPTY_SLOT) break;                        // claimed empty slot
        cur = (prev > cur) ? prev : cur;                      // carry displaced value down
    }
}

// ------------------------------- pass 4: coordinates + capped counts
__global__ void vox_finalize(const unsigned* __restrict__ counters,
                             const unsigned* __restrict__ linOfRank,
                             const unsigned* __restrict__ totalp,
                             float* __restrict__ coords,
                             float* __restrict__ counts) {
    int v = blockIdx.x * blockDim.x + threadIdx.x;
    if (v >= MAX_VOX) return;
    unsigned total = *totalp;
    unsigned nreal = total < (unsigned)MAX_VOX ? total : (unsigned)MAX_VOX;
    float cx = 0.f, cy = 0.f, cz = 0.f;
    if ((unsigned)v < nreal) {
        unsigned lin = linOfRank[v];
        unsigned x = lin / (unsigned)(GRIDY * GRIDZ);
        unsigned rem = lin % (unsigned)(GRIDY * GRIDZ);
        cx = (float)x;
        cy = (float)(rem / (unsigned)GRIDZ);
        cz = (float)(rem % (unsigned)GRIDZ);
    }
    coords[v * 3 + 0] = cx;
    coords[v * 3 + 1] = cy;
    coords[v * 3 + 2] = cz;
    unsigned c = counters[v];
    if (c > (unsigned)MAX_PTS) c = (unsigned)MAX_PTS;
    counts[v] = (float)c;
}

// ------------------------------- pass 5: gather points into voxel tensor
__global__ void vox_gather(const float4* __restrict__ pts,
                           const unsigned* __restrict__ slots,
                           float4* __restrict__ voxels) {
    int t = blockIdx.x * blockDim.x + threadIdx.x;
    if (t >= MAX_VOX * MAX_PTS) return;
    unsigned pi = slots[t];
    if (pi == EMPTY_SLOT) return;                             // stays zero
    voxels[t] = pts[pi];                                      // b128 load + b128 store
}

extern "C" void kernel_launch(void* const* d_in, const int* in_sizes, int n_in,
                              void* d_out, int out_size, void* d_ws, size_t ws_size,
                              hipStream_t stream) {
    (void)n_in; (void)out_size; (void)ws_size;
    const float4* pts = (const float4*)d_in[0];
    const int n = in_sizes[0] / 4;                            // 2,000,000 points

    unsigned* ws = (unsigned*)d_ws;
    unsigned* bitmap    = ws + OFF_BITMAP;
    unsigned* wpre      = ws + OFF_WPRE;
    unsigned* bsum      = ws + OFF_BSUM;
    unsigned* counters  = ws + OFF_COUNT;
    unsigned* linOfRank = ws + OFF_LINR;
    unsigned* slots     = ws + OFF_SLOT;
    unsigned* totalp    = ws + OFF_TOTAL;
    unsigned* plin      = ws + OFF_PLIN;

    float* out    = (float*)d_out;
    float* voxels = out;                                      // [120000][32][4] f32
    float* coords = out + (size_t)MAX_VOX * MAX_PTS * 4;      // [120000][3]
    float* counts = coords + (size_t)MAX_VOX * 3;             // [120000]

    const int BS = 256;

    // init (every call -> deterministic across graph replays); all regions /16
    vox_memset_u128<<<1024, BS, 0, stream>>>((uint4*)bitmap, 0u, (unsigned)(NWORDS / 4));
    vox_memset_u128<<<128,  BS, 0, stream>>>((uint4*)counters, 0u, (unsigned)(MAX_VOX / 4));
    vox_memset_u128<<<1024, BS, 0, stream>>>((uint4*)slots, EMPTY_SLOT, (unsigned)(MAX_VOX * MAX_PTS / 4));
    vox_memset_u128<<<2048, BS, 0, stream>>>((uint4*)voxels, 0u, (unsigned)(MAX_VOX * MAX_PTS));

    // pass 1: presence bitmap + cached per-point lin
    vox_mark_kernel<<<(n + BS - 1) / BS, BS, 0, stream>>>(pts, bitmap, plin, n);

    // pass 2: prefix scan of bitmap popcounts (rank structure)
    vox_scan_block<<<NBLK, SCAN_BS, 0, stream>>>(bitmap, wpre, bsum);
    vox_scan_bsum<<<1, SCAN_BS, 0, stream>>>(bsum, totalp, NBLK);
    vox_add_offsets<<<(NWORDS + BS - 1) / BS, BS, 0, stream>>>(wpre, bsum);

    // pass 3: per-point rank + deterministic slot insertion
    vox_insert_kernel<<<(n + BS - 1) / BS, BS, 0, stream>>>(
        plin, bitmap, wpre, counters, linOfRank, slots, n);

    // pass 4: coordinates + counts
    vox_finalize<<<(MAX_VOX + BS - 1) / BS, BS, 0, stream>>>(
        counters, linOfRank, totalp, coords, counts);

    // pass 5: gather points
    vox_gather<<<(MAX_VOX * MAX_PTS + BS - 1) / BS, BS, 0, stream>>>(
        pts, slots, (float4*)voxels);
}